// MeshFunctionalMapper_6408091206159
// MI455X (gfx1250) — compile-verified
//
#include <hip/hip_runtime.h>
#include <hip/hip_bf16.h>

typedef float v2f __attribute__((ext_vector_type(2)));
typedef float v8f __attribute__((ext_vector_type(8)));

#define BATCH 4
#define NPTS  50000
#define DDIM  128
#define KDIM  128
#define CHUNK 512
#define NCHUNKS ((NPTS + CHUNK - 1) / CHUNK)   // 98
#define LAMBDA 1e-3f

// ---------------------------------------------------------------------------
// Zero the accumulation buffers (A and Bm partial sums) in workspace.
// ---------------------------------------------------------------------------
__global__ __launch_bounds__(256) void zero_kernel(float* __restrict__ ws, int n) {
    int idx = blockIdx.x * 256 + threadIdx.x;
    if (idx < n) ws[idx] = 0.0f;
}

// ---------------------------------------------------------------------------
// Projection: A[b] = phi_M[b]^T @ M_F[b], Bm[b] = phi_N[b]^T @ N_G[b]
// Each workgroup (8 waves, 256 threads) handles one (batch, matrix, N-chunk).
// Wave w owns output row-block k0 = 16*w; 8 col tiles of 16 -> 8 v8f accs.
// Reduction via V_WMMA_F32_16X16X4_F32, partials combined with f32 atomics.
// ---------------------------------------------------------------------------
__global__ __launch_bounds__(256) void proj_kernel(
    const float* __restrict__ M_F, const float* __restrict__ N_G,
    const float* __restrict__ phi_M, const float* __restrict__ phi_N,
    float* __restrict__ A_ws, float* __restrict__ B_ws) {

    int bid   = blockIdx.x;
    int b     = bid / (2 * NCHUNKS);
    int rem   = bid - b * 2 * NCHUNKS;
    int mat   = rem / NCHUNKS;
    int chunk = rem - mat * NCHUNKS;

    const float* phi = (mat == 0 ? phi_M : phi_N) + (size_t)b * NPTS * KDIM;
    const float* dat = (mat == 0 ? M_F  : N_G ) + (size_t)b * NPTS * DDIM;
    float*       out = (mat == 0 ? A_ws : B_ws) + (size_t)b * KDIM * DDIM;

    int wave   = threadIdx.x >> 5;
    int lane   = threadIdx.x & 31;
    int lane15 = lane & 15;
    int hi     = lane >> 4;        // lane-half selects K pair {0,1} vs {2,3}
    int k0     = wave * 16;        // output row block

    int n_start = chunk * CHUNK;
    int n_end   = min(n_start + CHUNK, NPTS);

    v8f acc[8];
    for (int t = 0; t < 8; ++t) acc[t] = (v8f){0,0,0,0,0,0,0,0};

    for (int n = n_start; n < n_end; n += 4) {
        int r0 = n + 2 * hi;       // first reduction row for this lane-half
        // A-frag: element (m, kk) = phi[n+kk, k0+m]
        v2f a;
        a.x = phi[(size_t)r0       * KDIM + k0 + lane15];
        a.y = phi[(size_t)(r0 + 1) * KDIM + k0 + lane15];

        if (n + 32 < n_end)        // speculative stream-ahead prefetch
            __builtin_prefetch(&dat[(size_t)(n + 32) * DDIM + lane * 16], 0, 1);

        for (int t = 0; t < 8; ++t) {
            // B-frag: element (kk, nn) = dat[n+kk, 16*t+nn]
            v2f bf;
            bf.x = dat[(size_t)r0       * DDIM + t * 16 + lane15];
            bf.y = dat[(size_t)(r0 + 1) * DDIM + t * 16 + lane15];
            acc[t] = __builtin_amdgcn_wmma_f32_16x16x4_f32(
                false, a, false, bf, (short)0, acc[t], false, false);
        }
    }

    // C/D layout: element (m = v + 8*hi, n = lane15) in VGPR v
    for (int t = 0; t < 8; ++t)
        for (int v = 0; v < 8; ++v)
            atomicAdd(&out[(size_t)(k0 + v + 8 * hi) * DDIM + t * 16 + lane15],
                      acc[t][v]);
}

// ---------------------------------------------------------------------------
// Gram: AAt[b] = A[b] @ A[b]^T ; BAt[b] = Bm[b] @ A[b]^T   (reduce over D)
// One workgroup per (batch, matrix). Same WMMA tiling as above.
// ---------------------------------------------------------------------------
__global__ __launch_bounds__(256) void gram_kernel(
    const float* __restrict__ A_ws, const float* __restrict__ B_ws,
    float* __restrict__ AAt, float* __restrict__ BAt) {

    int bid = blockIdx.x;
    int b   = bid >> 1;
    int mat = bid & 1;

    const float* L  = (mat == 0 ? A_ws : B_ws) + (size_t)b * KDIM * DDIM;
    const float* R  = A_ws + (size_t)b * KDIM * DDIM;
    float*       out = (mat == 0 ? AAt : BAt) + (size_t)b * KDIM * KDIM;

    int wave   = threadIdx.x >> 5;
    int lane   = threadIdx.x & 31;
    int lane15 = lane & 15;
    int hi     = lane >> 4;
    int k0     = wave * 16;

    v8f acc[8];
    for (int t = 0; t < 8; ++t) acc[t] = (v8f){0,0,0,0,0,0,0,0};

    for (int d = 0; d < DDIM; d += 4) {
        int c0 = d + 2 * hi;
        // A-frag: element (m, kk) = L[k0+m, d+kk]
        v2f a;
        a.x = L[(size_t)(k0 + lane15) * DDIM + c0];
        a.y = L[(size_t)(k0 + lane15) * DDIM + c0 + 1];
        for (int t = 0; t < 8; ++t) {
            // B-frag: element (kk, l) = R[16*t+l, d+kk]  (i.e. R^T)
            v2f bf;
            bf.x = R[(size_t)(t * 16 + lane15) * DDIM + c0];
            bf.y = R[(size_t)(t * 16 + lane15) * DDIM + c0 + 1];
            acc[t] = __builtin_amdgcn_wmma_f32_16x16x4_f32(
                false, a, false, bf, (short)0, acc[t], false, false);
        }
    }

    for (int t = 0; t < 8; ++t)
        for (int v = 0; v < 8; ++v)
            out[(size_t)(k0 + v + 8 * hi) * KDIM + t * 16 + lane15] = acc[t][v];
}

// ---------------------------------------------------------------------------
// Solve: for each (b, i): (AAt[b] + 1e-3 * diag((lamM[b,:]-lamN[b,i])^2)) x =
// BAt[b,i,:].  SPD -> Gaussian elimination w/o pivoting, matrix in LDS.
// 128 threads: thread c owns column c (conflict-free LDS, broadcast factors).
// Dynamic LDS = 128*128 + 2*128 floats = 66560 B (CDNA5 LDS = 320 KB/WGP).
// ---------------------------------------------------------------------------
__global__ __launch_bounds__(128) void solve_kernel(
    const float* __restrict__ AAt, const float* __restrict__ BAt,
    const float* __restrict__ lamM, const float* __restrict__ lamN,
    float* __restrict__ C) {

    extern __shared__ float sm[];
    float* Msh = sm;                    // 128 x 128
    float* rhs = sm + KDIM * KDIM;      // 128
    float* xv  = rhs + KDIM;            // 128

    int bid = blockIdx.x;
    int b   = bid >> 7;
    int i   = bid & 127;
    int c   = threadIdx.x;

    const float* G = AAt + (size_t)b * KDIM * KDIM;
    for (int idx = c; idx < KDIM * KDIM; idx += 128) Msh[idx] = G[idx];
    rhs[c] = BAt[(size_t)b * KDIM * KDIM + (size_t)i * KDIM + c];
    __syncthreads();

    // add lambda * (lamM[b,c] - lamN[b,i])^2 on the diagonal
    float dl = lamM[b * KDIM + c] - lamN[b * KDIM + i];
    Msh[c * KDIM + c] += LAMBDA * dl * dl;
    __syncthreads();

    // forward elimination (no pivoting; SPD)
    for (int j = 0; j < KDIM - 1; ++j) {
        float pivinv = 1.0f / Msh[j * KDIM + j];
        float mjc = Msh[j * KDIM + c];   // pivot-row entry for my column
        float rj  = rhs[j];
        if (c > j) {
            for (int r = j + 1; r < KDIM; ++r) {
                float f = Msh[r * KDIM + j] * pivinv;  // LDS broadcast
                Msh[r * KDIM + c] -= f * mjc;
            }
        } else if (c == j) {
            for (int r = j + 1; r < KDIM; ++r) {
                float f = Msh[r * KDIM + j] * pivinv;
                rhs[r] -= f * rj;
            }
        }
        __syncthreads();
    }

    // back substitution
    for (int j = KDIM - 1; j >= 0; --j) {
        __syncthreads();
        float xj = rhs[j] / Msh[j * KDIM + j];
        if (c == j) xv[j] = xj;
        if (c < j)  rhs[c] -= Msh[c * KDIM + j] * xj;
    }
    __syncthreads();

    C[(size_t)b * KDIM * KDIM + (size_t)i * KDIM + c] = xv[c];
}

// ---------------------------------------------------------------------------
extern "C" void kernel_launch(void* const* d_in, const int* in_sizes, int n_in,
                              void* d_out, int out_size, void* d_ws, size_t ws_size,
                              hipStream_t stream) {
    const float* M_F   = (const float*)d_in[0];
    const float* N_G   = (const float*)d_in[1];
    const float* phi_M = (const float*)d_in[2];
    const float* phi_N = (const float*)d_in[3];
    const float* lamM  = (const float*)d_in[4];
    const float* lamN  = (const float*)d_in[5];
    float* C = (float*)d_out;

    float* ws   = (float*)d_ws;
    float* A_ws = ws;                 // 4*128*128
    float* B_ws = ws + 65536;         // 4*128*128
    float* AAt  = ws + 131072;        // 4*128*128
    float* BAt  = ws + 196608;        // 4*128*128

    // allow >64KB dynamic LDS for the solver (CDNA5: 320 KB/WGP)
    int smem_bytes = (KDIM * KDIM + 2 * KDIM) * (int)sizeof(float);
    hipFuncSetAttribute((const void*)solve_kernel,
                        hipFuncAttributeMaxDynamicSharedMemorySize, smem_bytes);

    zero_kernel<<<(131072 + 255) / 256, 256, 0, stream>>>(ws, 131072);
    proj_kernel<<<BATCH * 2 * NCHUNKS, 256, 0, stream>>>(M_F, N_G, phi_M, phi_N,
                                                         A_ws, B_ws);
    gram_kernel<<<BATCH * 2, 256, 0, stream>>>(A_ws, B_ws, AAt, BAt);
    solve_kernel<<<BATCH * KDIM, 128, smem_bytes, stream>>>(AAt, BAt, lamM, lamN, C);
}